// MaskLayer_48954037240117
// MI455X (gfx1250) — compile-verified
//
#include <hip/hip_runtime.h>
#include <math.h>

#define FB 64          // features per block tile
#define RPB 16         // rows (b,s) per block
#define TPB 256        // threads per block

typedef int v4i __attribute__((ext_vector_type(4)));
typedef __attribute__((address_space(1))) v4i gv4i;   // global int4
typedef __attribute__((address_space(3))) v4i lv4i;   // LDS int4
typedef __attribute__((address_space(3))) void lds_void;

__device__ __forceinline__ void async_load_b128(const float* gsrc, float* ldst) {
#if __has_builtin(__builtin_amdgcn_global_load_async_to_lds_b128)
    __builtin_amdgcn_global_load_async_to_lds_b128(
        (gv4i*)gsrc, (lv4i*)ldst, 0, 0);
#else
    unsigned l = (unsigned)(unsigned long long)(lds_void*)ldst;
    unsigned long long g = (unsigned long long)gsrc;
    asm volatile("global_load_async_to_lds_b128 %0, %1, off"
                 :: "v"(l), "v"(g) : "memory");
#endif
}

__device__ __forceinline__ void wait_async0() {
#if __has_builtin(__builtin_amdgcn_s_wait_asynccnt)
    __builtin_amdgcn_s_wait_asynccnt(0);
#else
    asm volatile("s_wait_asynccnt 0" ::: "memory");
#endif
}

// x: [BS, F]  kde: [N, F]  repl: [F]  out: [BS, F]
// grid = (F/FB, BS/RPB), block = TPB
__global__ void __launch_bounds__(TPB)
kde_mask_kernel(const float* __restrict__ x,
                const float* __restrict__ kde,
                const float* __restrict__ repl,
                float* __restrict__ out,
                int F, int N)
{
    __shared__ float s_tile[128 * FB];   // [n][f], pre-scaled kde tile (<=128 samples)
    __shared__ float s_scale[FB];        // sqrt(0.5*log2e)/bw
    __shared__ float s_thresh[FB];       // 0.3*N*bw/INV_SQRT_2PI

    const int t    = threadIdx.x;
    const int f    = t & (FB - 1);
    const int trow = t >> 6;             // 0..3
    const int f0   = blockIdx.x * FB;
    const int rowB = blockIdx.y * RPB;
    const int fg   = f0 + f;

    // ---- Phase 1: kick off async global->LDS fill of the raw kde tile ----
    // tile = N rows x FB floats; b128 chunks: (N*FB/4) total, TPB threads
    {
        const float* gbase = kde + f0;
        const int nchunks = (N * FB) >> 2;             // 2048 for N=128
        for (int c = t; c < nchunks; c += TPB) {
            int n  = c >> 4;                           // FB/4 = 16 chunks per row
            int fi = (c & 15) << 2;
            async_load_b128(gbase + n * F + fi, &s_tile[c << 2]);
        }
    }

    // ---- Phase 2 (overlapped): per-feature bandwidth stats, threads 0..FB-1 ----
    if (t < FB) {
        const float* col = kde + f0 + t;
        float sum = 0.f;
        for (int n = 0; n < N; ++n) sum += col[n * F];
        float mean = sum / (float)N;
        float ssd = 0.f;
        for (int n = 0; n < N; ++n) { float d = col[n * F] - mean; ssd += d * d; }
        float stdv = sqrtf(ssd / (float)(N - 1));
        float bw   = powf((float)N, -0.2f) * stdv;     // Scott's rule, d=1
        // exp(-0.5 z^2) == exp2(-(s*(x-d))^2) with s = sqrt(0.5*log2(e))/bw
        s_scale[t]  = 0.84932180f / bw;
        // mask iff sum_n exp(...) >= 0.3 * N * bw / INV_SQRT_2PI
        s_thresh[t] = (0.3f * (float)N / 0.3989422804014327f) * bw;
    }

    wait_async0();
    __syncthreads();

    // ---- Phase 3: scale tile in place by per-feature s ----
    for (int i = t; i < N * FB; i += TPB)
        s_tile[i] *= s_scale[i & (FB - 1)];
    __syncthreads();

    // ---- Phase 4: main compute. 4 rows per thread share each LDS read ----
    const float sc = s_scale[f];
    const float th = s_thresh[f];
    const float rv = repl[fg];

    const int r0 = rowB + trow;                        // rows r0, r0+4, r0+8, r0+12
    const float x0 = x[(r0 +  0) * F + fg];
    const float x1 = x[(r0 +  4) * F + fg];
    const float x2 = x[(r0 +  8) * F + fg];
    const float x3 = x[(r0 + 12) * F + fg];
    const float a0 = x0 * sc, a1 = x1 * sc, a2 = x2 * sc, a3 = x3 * sc;

    float c0 = 0.f, c1 = 0.f, c2 = 0.f, c3 = 0.f;
#pragma unroll 4
    for (int n = 0; n < N; ++n) {
        float d = s_tile[n * FB + f];
        float u0 = a0 - d; c0 += __builtin_amdgcn_exp2f(-(u0 * u0));
        float u1 = a1 - d; c1 += __builtin_amdgcn_exp2f(-(u1 * u1));
        float u2 = a2 - d; c2 += __builtin_amdgcn_exp2f(-(u2 * u2));
        float u3 = a3 - d; c3 += __builtin_amdgcn_exp2f(-(u3 * u3));
    }

    out[(r0 +  0) * F + fg] = (c0 >= th) ? rv : x0;
    out[(r0 +  4) * F + fg] = (c1 >= th) ? rv : x1;
    out[(r0 +  8) * F + fg] = (c2 >= th) ? rv : x2;
    out[(r0 + 12) * F + fg] = (c3 >= th) ? rv : x3;
}

extern "C" void kernel_launch(void* const* d_in, const int* in_sizes, int n_in,
                              void* d_out, int out_size, void* d_ws, size_t ws_size,
                              hipStream_t stream) {
    (void)n_in; (void)d_ws; (void)ws_size;
    const float* x    = (const float*)d_in[0];
    const float* kde  = (const float*)d_in[1];
    const float* repl = (const float*)d_in[2];
    float* out        = (float*)d_out;

    const int F  = in_sizes[2];            // 512
    const int N  = in_sizes[1] / F;        // 128
    const int BS = in_sizes[0] / F;        // 2048

    dim3 grid(F / FB, BS / RPB);
    kde_mask_kernel<<<grid, TPB, 0, stream>>>(x, kde, repl, out, F, N);
}